// E2PN_87222195847618
// MI455X (gfx1250) — compile-verified
//
#include <hip/hip_runtime.h>

// ---------------------------------------------------------------------------
// KPConv-style encoder/decoder for MI455X (gfx1250, wave32).
// * GEMMs: V_WMMA_F32_16X16X4_F32 (fp32 in/out == reference precision).
//   One 64x16 tile per wave, 4 accumulators sharing one B fragment,
//   unguarded K-loops (all Ci % 4 == 0; split-A = two sequential loops).
// * Gathers (the roofline bottleneck): wave-per-query, float4 (b128) channel
//   groups, neighbor indices broadcast via wave32 shuffles.
// * Stem (Ci=1): rank-1 outer product + wave-reduction mean gather.
// ---------------------------------------------------------------------------

#define LEAK 0.1f

typedef __attribute__((ext_vector_type(2))) float v2f;
typedef __attribute__((ext_vector_type(4))) float v4f;
typedef __attribute__((ext_vector_type(8))) float v8f;

__device__ __forceinline__ float lrelu_f(float x) { return x > 0.f ? x : LEAK * x; }

#define WMMA_F32(A_, B_, C_)                                                   \
  __builtin_amdgcn_wmma_f32_16x16x4_f32(false, (A_), false, (B_), (short)0,    \
                                        (C_), false, false)

// ---------------------------------------------------------------------------
// wave-per-query gathers; K == 32 neighbors, C % 4 == 0.
// lane k holds idx[q*32+k]; broadcast with shuffles; channels in float4 groups.
// ---------------------------------------------------------------------------
__global__ void gather_mean_k4(const float* __restrict__ x, const int* __restrict__ idx,
                               float* __restrict__ out, int Nq, int Ns, int C) {
  const int lane = threadIdx.x & 31;
  const int wid  = threadIdx.x >> 5;
  const long q = (long)blockIdx.x * (blockDim.x >> 5) + wid;
  if (q >= Nq) return;
  const int jl = idx[q * 32 + lane];
  const int C4 = C >> 2;
  for (int c4 = lane; c4 < C4; c4 += 32) {
    v4f s = {0.f, 0.f, 0.f, 0.f};
#pragma unroll
    for (int k = 0; k < 32; ++k) {
      const int j = __shfl(jl, k, 32);
      if (j < Ns) s += *((const v4f*)(x + (long)j * C) + c4);
    }
    *((v4f*)(out + q * C) + c4) = s * 0.03125f;  // 1/32
  }
}

__global__ void gather_max_k4(const float* __restrict__ x, const int* __restrict__ idx,
                              float* __restrict__ out, int Nq, int Ns, int C) {
  const int lane = threadIdx.x & 31;
  const int wid  = threadIdx.x >> 5;
  const long q = (long)blockIdx.x * (blockDim.x >> 5) + wid;
  if (q >= Nq) return;
  const int jl = idx[q * 32 + lane];
  const int C4 = C >> 2;
  for (int c4 = lane; c4 < C4; c4 += 32) {
    v4f s = {-__builtin_huge_valf(), -__builtin_huge_valf(),
             -__builtin_huge_valf(), -__builtin_huge_valf()};
#pragma unroll
    for (int k = 0; k < 32; ++k) {
      const int j = __shfl(jl, k, 32);
      v4f v = {0.f, 0.f, 0.f, 0.f};                 // shadow row
      if (j < Ns) v = *((const v4f*)(x + (long)j * C) + c4);
#pragma unroll
      for (int i = 0; i < 4; ++i) s[i] = fmaxf(s[i], v[i]);
    }
    *((v4f*)(out + q * C) + c4) = s;
  }
}

// nearest-neighbor upsample: out[q,:] = x[idx[q*32], :]
__global__ void gather_first_k4(const float* __restrict__ x, const int* __restrict__ idx,
                                float* __restrict__ out, int Nq, int Ns, int C) {
  const int lane = threadIdx.x & 31;
  const int wid  = threadIdx.x >> 5;
  const long q = (long)blockIdx.x * (blockDim.x >> 5) + wid;
  if (q >= Nq) return;
  const int j = idx[q * 32];
  const int C4 = C >> 2;
  const v4f zero = {0.f, 0.f, 0.f, 0.f};
  for (int c4 = lane; c4 < C4; c4 += 32)
    *((v4f*)(out + q * C) + c4) =
        (j < Ns) ? *((const v4f*)(x + (long)j * C) + c4) : zero;
}

// stem gather, C == 1: lane k loads one neighbor value, wave tree-reduce.
__global__ void gather_mean1_k(const float* __restrict__ x, const int* __restrict__ idx,
                               float* __restrict__ out, int Nq, int Ns) {
  const int lane = threadIdx.x & 31;
  const int wid  = threadIdx.x >> 5;
  const long q = (long)blockIdx.x * (blockDim.x >> 5) + wid;
  if (q >= Nq) return;
  const int j = idx[q * 32 + lane];
  float v = (j < Ns) ? x[j] : 0.f;
#pragma unroll
  for (int o = 16; o >= 1; o >>= 1) v += __shfl_xor(v, o, 32);
  if (lane == 0) out[q] = v * 0.03125f;
}

// out = lrelu(a + b)
__global__ void add_lrelu_k(const float* __restrict__ a, const float* __restrict__ b,
                            float* __restrict__ out, long n) {
  long i = (long)blockIdx.x * blockDim.x + threadIdx.x;
  const long stride = (long)gridDim.x * blockDim.x;
  for (; i < n; i += stride) out[i] = lrelu_f(a[i] + b[i]);
}

// stem GEMM: out[m, c] = lrelu(g[m] * w[c])   (rank-1 outer product, Ci == 1)
__global__ void outer_lrelu_k(const float* __restrict__ g, const float* __restrict__ w,
                              float* __restrict__ out, int M, int Co) {
  const long n = (long)M * Co;
  long i = (long)blockIdx.x * blockDim.x + threadIdx.x;
  const long stride = (long)gridDim.x * blockDim.x;
  for (; i < n; i += stride) {
    const int m = (int)(i / Co);
    const int c = (int)(i % Co);
    out[i] = lrelu_f(g[m] * w[c]);
  }
}

// ---------------------------------------------------------------------------
// fp32 WMMA GEMM:  Out[M,Co] = act( [A1|A2][M,C1+C2] @ W[Ci,Co]  (+ R) )
//   * one 64x16 tile per wave (4 accumulators share one B fragment)
//   * requires M % 64 == 0, Co % 16 == 0, C1 % 4 == 0, C2 % 4 == 0
//   * A frag (16x4): lane = M + 16*(K>>1), vgpr = K&1  (ISA 7.12.2)
//   * B frag (4x16): lane = N + 16*(K>>1), vgpr = K&1  (mirrored)
//   * C/D (16x16):   lane = N + 16*hi, vgpr v -> row v + 8*hi
// ---------------------------------------------------------------------------
__global__ __launch_bounds__(256) void wmma_gemm_k(
    const float* __restrict__ A1, int C1,
    const float* __restrict__ A2, int C2,
    const float* __restrict__ W,
    const float* __restrict__ R,
    float* __restrict__ Out,
    int M, int Co, int act) {
  const int lane = threadIdx.x & 31;
  const int nrow = lane & 15;       // M within sub-tile for A; N within tile for B/C/D
  const int hi   = lane >> 4;       // which K pair (A/B) / row half (C/D)
  const int wavesPerBlock = blockDim.x >> 5;
  const int waveId = blockIdx.x * wavesPerBlock + (threadIdx.x >> 5);
  const int nWaves = gridDim.x * wavesPerBlock;
  const int tilesN = Co >> 4;
  const long nTiles = (long)(M >> 6) * tilesN;

  for (long t = waveId; t < nTiles; t += nWaves) {
    const int bm = (int)(t / tilesN);
    const int tn = (int)(t % tilesN);
    const int m0 = bm << 6;
    const int n0 = tn << 4;
    const int bcol = n0 + nrow;

    v8f acc0 = {0.f, 0.f, 0.f, 0.f, 0.f, 0.f, 0.f, 0.f};
    v8f acc1 = acc0, acc2 = acc0, acc3 = acc0;

    // ---- K-loop over A1 (unguarded: C1 % 4 == 0) ----
    {
      const float* pA = A1 + (long)(m0 + nrow) * C1 + 2 * hi;
      const float* pW = W + (long)(2 * hi) * Co + bcol;
      const long sA = (long)16 * C1;
      const long sW = (long)4 * Co;
      for (int k0 = 0; k0 < C1; k0 += 4) {
        const v2f a0 = *(const v2f*)(pA);
        const v2f a1 = *(const v2f*)(pA + sA);
        const v2f a2 = *(const v2f*)(pA + 2 * sA);
        const v2f a3 = *(const v2f*)(pA + 3 * sA);
        v2f b; b.x = pW[0]; b.y = pW[Co];
        acc0 = WMMA_F32(a0, b, acc0);
        acc1 = WMMA_F32(a1, b, acc1);
        acc2 = WMMA_F32(a2, b, acc2);
        acc3 = WMMA_F32(a3, b, acc3);
        pA += 4; pW += sW;
      }
    }
    // ---- K-loop over A2 (W rows offset by C1; unguarded: C2 % 4 == 0) ----
    if (C2 > 0) {
      const float* pA = A2 + (long)(m0 + nrow) * C2 + 2 * hi;
      const float* pW = W + (long)(C1 + 2 * hi) * Co + bcol;
      const long sA = (long)16 * C2;
      const long sW = (long)4 * Co;
      for (int k0 = 0; k0 < C2; k0 += 4) {
        const v2f a0 = *(const v2f*)(pA);
        const v2f a1 = *(const v2f*)(pA + sA);
        const v2f a2 = *(const v2f*)(pA + 2 * sA);
        const v2f a3 = *(const v2f*)(pA + 3 * sA);
        v2f b; b.x = pW[0]; b.y = pW[Co];
        acc0 = WMMA_F32(a0, b, acc0);
        acc1 = WMMA_F32(a1, b, acc1);
        acc2 = WMMA_F32(a2, b, acc2);
        acc3 = WMMA_F32(a3, b, acc3);
        pA += 4; pW += sW;
      }
    }

    // ---- epilogue: optional residual + activation ----
#pragma unroll
    for (int v = 0; v < 8; ++v) {
      const int r0 = m0 + v + 8 * hi;
      float x0 = acc0[v], x1 = acc1[v], x2 = acc2[v], x3 = acc3[v];
      if (R) {
        x0 += R[(long)(r0)      * Co + bcol];
        x1 += R[(long)(r0 + 16) * Co + bcol];
        x2 += R[(long)(r0 + 32) * Co + bcol];
        x3 += R[(long)(r0 + 48) * Co + bcol];
      }
      if (act) {
        x0 = lrelu_f(x0); x1 = lrelu_f(x1); x2 = lrelu_f(x2); x3 = lrelu_f(x3);
      }
      Out[(long)(r0)      * Co + bcol] = x0;
      Out[(long)(r0 + 16) * Co + bcol] = x1;
      Out[(long)(r0 + 32) * Co + bcol] = x2;
      Out[(long)(r0 + 48) * Co + bcol] = x3;
    }
  }
}

// ---------------------------------------------------------------------------
// Host-side orchestration
// ---------------------------------------------------------------------------
static inline void gemm2(hipStream_t s, const float* A1, int C1, const float* A2, int C2,
                         const float* W, const float* R, float* Out, int M, int Co, int act) {
  const long tiles = (long)(M / 64) * (Co / 16);
  int blocks = (int)((tiles + 7) / 8);
  if (blocks > 8192) blocks = 8192;
  if (blocks < 1) blocks = 1;
  wmma_gemm_k<<<blocks, 256, 0, s>>>(A1, C1, A2, C2, W, R, Out, M, Co, act);
}

static inline void gemm(hipStream_t s, const float* A, int Ci, const float* W,
                        const float* R, float* Out, int M, int Co, int act) {
  gemm2(s, A, Ci, nullptr, 0, W, R, Out, M, Co, act);
}

static inline void gmean(hipStream_t s, const float* x, const int* idx, float* out,
                         int Nq, int Ns, int C) {
  gather_mean_k4<<<(Nq + 7) / 8, 256, 0, s>>>(x, idx, out, Nq, Ns, C);
}
static inline void gmax(hipStream_t s, const float* x, const int* idx, float* out,
                        int Nq, int Ns, int C) {
  gather_max_k4<<<(Nq + 7) / 8, 256, 0, s>>>(x, idx, out, Nq, Ns, C);
}
static inline void gfirst(hipStream_t s, const float* x, const int* idx, float* out,
                          int Nq, int Ns, int C) {
  gather_first_k4<<<(Nq + 7) / 8, 256, 0, s>>>(x, idx, out, Nq, Ns, C);
}

static inline void addact(hipStream_t s, const float* a, const float* b, float* out, long n) {
  int blocks = (int)((n + 255) / 256);
  if (blocks > 16384) blocks = 16384;
  add_lrelu_k<<<blocks, 256, 0, s>>>(a, b, out, n);
}

extern "C" void kernel_launch(void* const* d_in, const int* in_sizes, int n_in,
                              void* d_out, int out_size, void* d_ws, size_t ws_size,
                              hipStream_t stream) {
  (void)in_sizes; (void)n_in; (void)out_size;
  const int N1 = 65536, N2 = 16384, N3 = 4096, N4 = 1024;

  // inputs in setup_inputs() dict order
  const float* feats = (const float*)d_in[0];
  const int *nbr1 = (const int*)d_in[1], *nbr2 = (const int*)d_in[2];
  const int *nbr3 = (const int*)d_in[3], *nbr4 = (const int*)d_in[4];
  const int *sub1 = (const int*)d_in[5], *sub2 = (const int*)d_in[6], *sub3 = (const int*)d_in[7];
  const int *up32 = (const int*)d_in[8], *up21 = (const int*)d_in[9];
  const float* w_e11   = (const float*)d_in[10];
  const float *e12_w1 = (const float*)d_in[11], *e12_wc = (const float*)d_in[12],
              *e12_w3 = (const float*)d_in[13], *e12_wsc = (const float*)d_in[14];
  const float *e21_w1 = (const float*)d_in[15], *e21_wc = (const float*)d_in[16],
              *e21_w3 = (const float*)d_in[17];
  const float *e22_w1 = (const float*)d_in[18], *e22_wc = (const float*)d_in[19],
              *e22_w3 = (const float*)d_in[20], *e22_wsc = (const float*)d_in[21];
  const float *e23_w1 = (const float*)d_in[22], *e23_wc = (const float*)d_in[23],
              *e23_w3 = (const float*)d_in[24];
  const float *e31_w1 = (const float*)d_in[25], *e31_wc = (const float*)d_in[26],
              *e31_w3 = (const float*)d_in[27];
  const float *e32_w1 = (const float*)d_in[28], *e32_wc = (const float*)d_in[29],
              *e32_w3 = (const float*)d_in[30], *e32_wsc = (const float*)d_in[31];
  const float *e33_w1 = (const float*)d_in[32], *e33_wc = (const float*)d_in[33],
              *e33_w3 = (const float*)d_in[34];
  const float *e41_w1 = (const float*)d_in[35], *e41_wc = (const float*)d_in[36],
              *e41_w3 = (const float*)d_in[37];
  const float *e42_w1 = (const float*)d_in[38], *e42_wc = (const float*)d_in[39],
              *e42_w3 = (const float*)d_in[40], *e42_wsc = (const float*)d_in[41];
  const float *e43_w1 = (const float*)d_in[42], *e43_wc = (const float*)d_in[43],
              *e43_w3 = (const float*)d_in[44];
  const float *wd3 = (const float*)d_in[45], *wd2 = (const float*)d_in[46];

  // outputs: (l2, l3, s4) flattened in return order
  float* out    = (float*)d_out;
  float* l2_out = out;                               // [N2, 256]
  float* l3_out = l2_out + (long)N2 * 256;           // [N3, 512]
  float* s4_out = l3_out + (long)N3 * 512;           // [N4, 1024]

  // workspace layout (floats): 3 ping-pong big buffers + 2 persistent skips
  const long SZ_BIG = (long)N1 * 128;                // 8,388,608
  const long SZ_T   = (long)N2 * 256;                // 4,194,304
  const size_t need = (size_t)(3 * SZ_BIG + 2 * SZ_T) * sizeof(float); // 128 MiB
  if (ws_size < need) return;
  float* B0 = (float*)d_ws;
  float* B1 = B0 + SZ_BIG;
  float* B2 = B1 + SZ_BIG;
  float* T0 = B2 + SZ_BIG;                           // s2 [N2,256]
  float* T1 = T0 + SZ_T;                             // s3 [N3,512]

  // ---- stem: s1a = lrelu(mean-gather(feats) @ w_e11)  [N1,64] -> T0 ----
  gather_mean1_k<<<(N1 + 7) / 8, 256, 0, stream>>>(feats, nbr1, B1, N1, N1);
  outer_lrelu_k<<<16384, 256, 0, stream>>>(B1, w_e11, T0, N1, 64);

  // ---- e12 (resnet, wsc): s1 [N1,128] -> B2 ----
  gemm(stream, T0, 64, e12_w1, nullptr, B0, N1, 32, 1);
  gmean(stream, B0, nbr1, B1, N1, N1, 32);
  gemm(stream, B1, 32, e12_wc, nullptr, B0, N1, 32, 1);
  gemm(stream, B0, 32, e12_w3, nullptr, B1, N1, 128, 0);
  gemm(stream, T0, 64, e12_wsc, B1, B2, N1, 128, 1);

  // ---- e21 (strided): s2a [N2,128] -> B2 ----
  gemm(stream, B2, 128, e21_w1, nullptr, B0, N1, 32, 1);
  gmean(stream, B0, sub1, B1, N2, N1, 32);
  gemm(stream, B1, 32, e21_wc, nullptr, B0, N2, 32, 1);
  gemm(stream, B0, 32, e21_w3, nullptr, B1, N2, 128, 0);
  gmax(stream, B2, sub1, B0, N2, N1, 128);
  addact(stream, B1, B0, B2, (long)N2 * 128);

  // ---- e22 (wsc): s2b [N2,256] -> B0 ----
  gemm(stream, B2, 128, e22_w1, nullptr, B0, N2, 64, 1);
  gmean(stream, B0, nbr2, B1, N2, N2, 64);
  gemm(stream, B1, 64, e22_wc, nullptr, B0, N2, 64, 1);
  gemm(stream, B0, 64, e22_w3, nullptr, B1, N2, 256, 0);
  gemm(stream, B2, 128, e22_wsc, B1, B0, N2, 256, 1);

  // ---- e23: s2 [N2,256] -> T0 ----
  gemm(stream, B0, 256, e23_w1, nullptr, B1, N2, 64, 1);
  gmean(stream, B1, nbr2, B2, N2, N2, 64);
  gemm(stream, B2, 64, e23_wc, nullptr, B1, N2, 64, 1);
  gemm(stream, B1, 64, e23_w3, nullptr, B2, N2, 256, 0);
  addact(stream, B2, B0, T0, (long)N2 * 256);

  // ---- e31 (strided): s3a [N3,256] -> B2 ----
  gemm(stream, T0, 256, e31_w1, nullptr, B0, N2, 64, 1);
  gmean(stream, B0, sub2, B1, N3, N2, 64);
  gemm(stream, B1, 64, e31_wc, nullptr, B0, N3, 64, 1);
  gemm(stream, B0, 64, e31_w3, nullptr, B1, N3, 256, 0);
  gmax(stream, T0, sub2, B0, N3, N2, 256);
  addact(stream, B1, B0, B2, (long)N3 * 256);

  // ---- e32 (wsc): s3b [N3,512] -> B0 ----
  gemm(stream, B2, 256, e32_w1, nullptr, B0, N3, 128, 1);
  gmean(stream, B0, nbr3, B1, N3, N3, 128);
  gemm(stream, B1, 128, e32_wc, nullptr, B0, N3, 128, 1);
  gemm(stream, B0, 128, e32_w3, nullptr, B1, N3, 512, 0);
  gemm(stream, B2, 256, e32_wsc, B1, B0, N3, 512, 1);

  // ---- e33: s3 [N3,512] -> T1 ----
  gemm(stream, B0, 512, e33_w1, nullptr, B1, N3, 128, 1);
  gmean(stream, B1, nbr3, B2, N3, N3, 128);
  gemm(stream, B2, 128, e33_wc, nullptr, B1, N3, 128, 1);
  gemm(stream, B1, 128, e33_w3, nullptr, B2, N3, 512, 0);
  addact(stream, B2, B0, T1, (long)N3 * 512);

  // ---- e41 (strided): s4a [N4,512] -> B2 ----
  gemm(stream, T1, 512, e41_w1, nullptr, B0, N3, 128, 1);
  gmean(stream, B0, sub3, B1, N4, N3, 128);
  gemm(stream, B1, 128, e41_wc, nullptr, B0, N4, 128, 1);
  gemm(stream, B0, 128, e41_w3, nullptr, B1, N4, 512, 0);
  gmax(stream, T1, sub3, B0, N4, N3, 512);
  addact(stream, B1, B0, B2, (long)N4 * 512);

  // ---- e42 (wsc): s4b [N4,1024] -> B0 ----
  gemm(stream, B2, 512, e42_w1, nullptr, B0, N4, 256, 1);
  gmean(stream, B0, nbr4, B1, N4, N4, 256);
  gemm(stream, B1, 256, e42_wc, nullptr, B0, N4, 256, 1);
  gemm(stream, B0, 256, e42_w3, nullptr, B1, N4, 1024, 0);
  gemm(stream, B2, 512, e42_wsc, B1, B0, N4, 1024, 1);

  // ---- e43: s4 [N4,1024] -> s4_out ----
  gemm(stream, B0, 1024, e43_w1, nullptr, B1, N4, 256, 1);
  gmean(stream, B1, nbr4, B2, N4, N4, 256);
  gemm(stream, B2, 256, e43_wc, nullptr, B1, N4, 256, 1);
  gemm(stream, B1, 256, e43_w3, nullptr, B2, N4, 1024, 0);
  addact(stream, B2, B0, s4_out, (long)N4 * 1024);

  // ---- decoder (split-A GEMMs avoid materialized concat) ----
  gfirst(stream, s4_out, up32, B0, N3, N4, 1024);
  gemm2(stream, B0, 1024, T1, 512, wd3, nullptr, l3_out, N3, 512, 1);   // l3 [N3,512]
  gfirst(stream, l3_out, up21, B0, N2, N3, 512);
  gemm2(stream, B0, 512, T0, 256, wd2, nullptr, l2_out, N2, 256, 0);    // l2 [N2,256]
}